// NetVLAD_27152783245666
// MI455X (gfx1250) — compile-verified
//
#include <hip/hip_runtime.h>

typedef __attribute__((ext_vector_type(2))) float v2f;
typedef __attribute__((ext_vector_type(8))) float v8f;

namespace {
constexpr int N_ = 64;     // batch
constexpr int C_ = 512;    // channels
constexpr int S_ = 1024;   // spatial H*W
constexpr int K_ = 64;     // clusters
constexpr float EPS_ = 1e-12f;
}

// D = A(16x4) * B(4x16) + C, f32 WMMA. Guarded so the host semantic pass
// (no amdgcn builtins) still compiles; device pass lowers to
// v_wmma_f32_16x16x4_f32.
__device__ __forceinline__ v8f wmma_f32_16x16x4(v2f a, v2f b, v8f c) {
#if __has_builtin(__builtin_amdgcn_wmma_f32_16x16x4_f32)
  return __builtin_amdgcn_wmma_f32_16x16x4_f32(
      /*neg_a=*/false, a, /*neg_b=*/false, b,
      /*c_mod=*/(short)0, c, /*reuse_a=*/false, /*reuse_b=*/false);
#else
  c[0] += a.x * b.x + a.y * b.y;  // host-pass placeholder only
  return c;
#endif
}

// ---------------------------------------------------------------------------
// 1) inv_norm[n,s] = 1 / max(||x[n,:,s]||_2, eps)
// ---------------------------------------------------------------------------
__global__ void k_invnorm(const float* __restrict__ x,
                          float* __restrict__ inv_norm) {
  int g = blockIdx.x * blockDim.x + threadIdx.x;  // 0 .. N*S-1
  int n = g >> 10;
  int s = g & (S_ - 1);
  const float* p = x + (size_t)n * C_ * S_ + s;
  float ss = 0.f;
#pragma unroll 4
  for (int c = 0; c < C_; ++c) {
    float v = p[(size_t)c * S_];
    ss += v * v;
  }
  inv_norm[g] = 1.0f / fmaxf(sqrtf(ss), EPS_);
}

// ---------------------------------------------------------------------------
// 2) logits[n,k,s] = sum_c W[k,c] * xhat[n,c,s] + b[k]
//    Block: one n, one 64-wide S tile. M=K=64, N=Stile=64, reduce C by 32.
// ---------------------------------------------------------------------------
__global__ void k_logits(const float* __restrict__ x,
                         const float* __restrict__ w,
                         const float* __restrict__ bias,
                         const float* __restrict__ inv_norm,
                         float* __restrict__ logits) {
  __shared__ float wt[64][36];  // [k][c] ; stride 36 -> conflict-free col reads
  __shared__ float xt[32][72];  // [c][s] ; stride 72 -> half-waves on disjoint banks
  const int n = blockIdx.x;
  const int s0 = blockIdx.y * 64;
  const int t = threadIdx.x;
  const int lane = t & 31, wv = t >> 5;
  const int hi = lane >> 4, lo = lane & 15;
  const int mt0 = wv >> 2, nt0 = wv & 3;
  const int mt1 = (wv + 8) >> 2, nt1 = (wv + 8) & 3;
  v8f acc0 = {}, acc1 = {};

  const float* xn = x + (size_t)n * C_ * S_;
  const float* inrm = inv_norm + n * S_;

  for (int c0 = 0; c0 < C_; c0 += 32) {
#pragma unroll
    for (int i = 0; i < 8; ++i) {  // stage W tile [64 k][32 c]
      int idx = t + i * 256;
      int r = idx >> 5, cc = idx & 31;
      wt[r][cc] = w[r * C_ + c0 + cc];
    }
#pragma unroll
    for (int i = 0; i < 8; ++i) {  // stage normalized X tile [32 c][64 s]
      int idx = t + i * 256;
      int cc = idx >> 6, sc = idx & 63;
      xt[cc][sc] = xn[(size_t)(c0 + cc) * S_ + s0 + sc] * inrm[s0 + sc];
      if (i == 0 && c0 + 32 < C_)
        __builtin_prefetch(&xn[(size_t)(c0 + 32 + cc) * S_ + s0 + sc], 0, 1);
    }
    __syncthreads();
#pragma unroll
    for (int cs = 0; cs < 32; cs += 4) {
      int ch = cs + 2 * hi;  // half-wave K split {0,1}/{2,3}
      v2f a0, b0, a1, b1;
      a0.x = wt[mt0 * 16 + lo][ch];
      a0.y = wt[mt0 * 16 + lo][ch + 1];
      b0.x = xt[ch][nt0 * 16 + lo];
      b0.y = xt[ch + 1][nt0 * 16 + lo];
      acc0 = wmma_f32_16x16x4(a0, b0, acc0);
      a1.x = wt[mt1 * 16 + lo][ch];
      a1.y = wt[mt1 * 16 + lo][ch + 1];
      b1.x = xt[ch][nt1 * 16 + lo];
      b1.y = xt[ch + 1][nt1 * 16 + lo];
      acc1 = wmma_f32_16x16x4(a1, b1, acc1);
    }
    __syncthreads();
  }

  float* lp = logits + (size_t)n * K_ * S_;
#pragma unroll
  for (int r = 0; r < 8; ++r) {  // C/D layout: VGPR r holds rows r / r+8
    int k0 = mt0 * 16 + r + 8 * hi;
    lp[(size_t)k0 * S_ + s0 + nt0 * 16 + lo] = acc0[r] + bias[k0];
    int k1 = mt1 * 16 + r + 8 * hi;
    lp[(size_t)k1 * S_ + s0 + nt1 * 16 + lo] = acc1[r] + bias[k1];
  }
}

// ---------------------------------------------------------------------------
// 3) softmax over the spatial dim (1024) per (n,k) row, in place.
// ---------------------------------------------------------------------------
__global__ void k_softmax(float* __restrict__ a, float* __restrict__ a_sum) {
  __shared__ float red[256];
  const int t = threadIdx.x;
  float* row = a + (size_t)blockIdx.x * S_;
  float v0 = row[t], v1 = row[t + 256], v2 = row[t + 512], v3 = row[t + 768];
  float m = fmaxf(fmaxf(v0, v1), fmaxf(v2, v3));
  red[t] = m;
  __syncthreads();
  for (int o = 128; o > 0; o >>= 1) {
    if (t < o) red[t] = fmaxf(red[t], red[t + o]);
    __syncthreads();
  }
  m = red[0];
  __syncthreads();
  float e0 = __expf(v0 - m), e1 = __expf(v1 - m);
  float e2 = __expf(v2 - m), e3 = __expf(v3 - m);
  red[t] = e0 + e1 + e2 + e3;
  __syncthreads();
  for (int o = 128; o > 0; o >>= 1) {
    if (t < o) red[t] += red[t + o];
    __syncthreads();
  }
  float inv = 1.0f / red[0];
  row[t] = e0 * inv;
  row[t + 256] = e1 * inv;
  row[t + 512] = e2 * inv;
  row[t + 768] = e3 * inv;
  if (t == 0) a_sum[blockIdx.x] = 1.0f;  // softmax rows sum to 1 by construction
}

// ---------------------------------------------------------------------------
// 4) vlad_raw[n,k,c] = sum_s a[n,k,s]*xhat[n,c,s] - a_sum[n,k]*cent[k,c]
//    Block: one n, one 64-wide C tile. M=K=64, N=Ctile=64, reduce S by 32.
// ---------------------------------------------------------------------------
__global__ void k_vlad(const float* __restrict__ x,
                       const float* __restrict__ a,
                       const float* __restrict__ inv_norm,
                       const float* __restrict__ a_sum,
                       const float* __restrict__ cent,
                       float* __restrict__ out) {
  __shared__ float at[64][36];  // [k][s]
  __shared__ float xt[32][72];  // [s][c] (transposed on store)
  const int n = blockIdx.x;
  const int c0 = blockIdx.y * 64;
  const int t = threadIdx.x;
  const int lane = t & 31, wv = t >> 5;
  const int hi = lane >> 4, lo = lane & 15;
  const int mt0 = wv >> 2, nt0 = wv & 3;
  const int mt1 = (wv + 8) >> 2, nt1 = (wv + 8) & 3;
  v8f acc0 = {}, acc1 = {};

  const float* an = a + (size_t)n * K_ * S_;
  const float* xn = x + (size_t)n * C_ * S_;
  const float* inrm = inv_norm + n * S_;

  for (int s0 = 0; s0 < S_; s0 += 32) {
#pragma unroll
    for (int i = 0; i < 8; ++i) {  // stage attention tile [64 k][32 s]
      int idx = t + i * 256;
      int r = idx >> 5, sc = idx & 31;
      at[r][sc] = an[(size_t)r * S_ + s0 + sc];
    }
#pragma unroll
    for (int i = 0; i < 8; ++i) {  // stage normalized X tile, transposed to [s][c]
      int idx = t + i * 256;
      int cc = idx >> 5, sc = idx & 31;  // coalesced along s
      xt[sc][cc] = xn[(size_t)(c0 + cc) * S_ + s0 + sc] * inrm[s0 + sc];
      if (i == 0 && s0 + 32 < S_)
        __builtin_prefetch(&xn[(size_t)(c0 + cc) * S_ + s0 + 32 + sc], 0, 1);
    }
    __syncthreads();
#pragma unroll
    for (int ss = 0; ss < 32; ss += 4) {
      int sh = ss + 2 * hi;
      v2f a0, b0, a1, b1;
      a0.x = at[mt0 * 16 + lo][sh];
      a0.y = at[mt0 * 16 + lo][sh + 1];
      b0.x = xt[sh][nt0 * 16 + lo];
      b0.y = xt[sh + 1][nt0 * 16 + lo];
      acc0 = wmma_f32_16x16x4(a0, b0, acc0);
      a1.x = at[mt1 * 16 + lo][sh];
      a1.y = at[mt1 * 16 + lo][sh + 1];
      b1.x = xt[sh][nt1 * 16 + lo];
      b1.y = xt[sh + 1][nt1 * 16 + lo];
      acc1 = wmma_f32_16x16x4(a1, b1, acc1);
    }
    __syncthreads();
  }

  float* on = out + (size_t)n * K_ * C_;
  const float* asn = a_sum + n * K_;
#pragma unroll
  for (int r = 0; r < 8; ++r) {
    int k0 = mt0 * 16 + r + 8 * hi;
    int ci0 = c0 + nt0 * 16 + lo;
    on[(size_t)k0 * C_ + ci0] = acc0[r] - asn[k0] * cent[(size_t)k0 * C_ + ci0];
    int k1 = mt1 * 16 + r + 8 * hi;
    int ci1 = c0 + nt1 * 16 + lo;
    on[(size_t)k1 * C_ + ci1] = acc1[r] - asn[k1] * cent[(size_t)k1 * C_ + ci1];
  }
}

// ---------------------------------------------------------------------------
// 5) intra-cluster L2 norm per (n,k) over C; accumulate post-norm energy per n
// ---------------------------------------------------------------------------
__global__ void k_intranorm(float* __restrict__ out, float* __restrict__ nsum) {
  __shared__ float red[256];
  const int t = threadIdx.x;
  float* base = out + (size_t)blockIdx.x * C_;
  float v0 = base[t], v1 = base[t + 256];
  red[t] = v0 * v0 + v1 * v1;
  __syncthreads();
  for (int o = 128; o > 0; o >>= 1) {
    if (t < o) red[t] += red[t + o];
    __syncthreads();
  }
  float total = red[0];
  float inv = 1.0f / fmaxf(sqrtf(total), EPS_);
  base[t] = v0 * inv;
  base[t + 256] = v1 * inv;
  if (t == 0) atomicAdd(&nsum[blockIdx.x >> 6], total * inv * inv);
}

// ---------------------------------------------------------------------------
// 6) final global L2 norm per n over K*C
// ---------------------------------------------------------------------------
__global__ void k_finalnorm(float* __restrict__ out,
                            const float* __restrict__ nsum) {
  int g = blockIdx.x * blockDim.x + threadIdx.x;
  int n = g >> 15;  // K*C = 32768
  float inv = 1.0f / fmaxf(sqrtf(nsum[n]), EPS_);
  out[g] *= inv;
}

__global__ void k_zero(float* __restrict__ p) { p[threadIdx.x] = 0.f; }

// ---------------------------------------------------------------------------
extern "C" void kernel_launch(void* const* d_in, const int* in_sizes, int n_in,
                              void* d_out, int out_size, void* d_ws,
                              size_t ws_size, hipStream_t stream) {
  (void)in_sizes; (void)n_in; (void)out_size; (void)ws_size;
  const float* x    = (const float*)d_in[0];  // [N,C,H,W]
  const float* w    = (const float*)d_in[1];  // [K,C]
  const float* bias = (const float*)d_in[2];  // [K]
  const float* cent = (const float*)d_in[3];  // [K,C]
  float* out = (float*)d_out;                 // [N, K*C]

  // workspace layout (~17.1 MB)
  float* inv_norm = (float*)d_ws;                       // N*S
  float* a_buf    = inv_norm + (size_t)N_ * S_;         // N*K*S (logits -> a)
  float* a_sum    = a_buf + (size_t)N_ * K_ * S_;       // N*K
  float* nsum     = a_sum + (size_t)N_ * K_;            // N

  k_zero<<<1, N_, 0, stream>>>(nsum);
  k_invnorm<<<(N_ * S_) / 256, 256, 0, stream>>>(x, inv_norm);
  k_logits<<<dim3(N_, S_ / 64), 256, 0, stream>>>(x, w, bias, inv_norm, a_buf);
  k_softmax<<<N_ * K_, 256, 0, stream>>>(a_buf, a_sum);
  k_vlad<<<dim3(N_, C_ / 64), 256, 0, stream>>>(x, a_buf, inv_norm, a_sum, cent, out);
  k_intranorm<<<N_ * K_, 256, 0, stream>>>(out, nsum);
  k_finalnorm<<<(N_ * K_ * C_) / 256, 256, 0, stream>>>(out, nsum);
}